// ScaleSpaceDetector_67113158967352
// MI455X (gfx1250) — compile-verified
//
#include <hip/hip_runtime.h>
#include <math.h>

// ---------------------------------------------------------------------------
// CDNA5 (gfx1250) scale-space Hessian detector.
//   - Gaussian blurs run on V_WMMA_F32_16X16X4_F32 (full f32 matrix pipe)
//   - conv patches staged with global_load_async_to_lds_b32 / s_wait_asynccnt
//   - per-octave top-2048 via u64 atomic-max buckets (L2 atomics)
//   - global top-2048 via one-workgroup 16K-entry LDS bitonic sort (128KB LDS)
// ---------------------------------------------------------------------------

typedef float v2f __attribute__((ext_vector_type(2)));
typedef float v8f __attribute__((ext_vector_type(8)));

#define NFEAT   2048
#define PLANE   (1024 * 1024)   // fixed plane stride, reused across octaves
#define NLVL    6
#define NOCT    7
#define NEG_BIG (-3.402823e38f)
#define TPAD    16              // zero-pad margin of the tap table

__device__ __constant__ float LVL_SIG[NLVL] = {
    1.6f, 2.01587368f, 2.53984168f, 3.2f, 4.03174736f, 5.07968337f};

__device__ __forceinline__ int clampi(int v, int lo, int hi) {
  return v < lo ? lo : (v > hi ? hi : v);
}
__device__ __forceinline__ int reflecti(int i, int n) {  // jnp 'reflect'
  if (i < 0) i = -i;
  if (i >= n) i = 2 * n - 2 - i;
  return i;
}
__device__ __forceinline__ unsigned fkey(float f) {  // monotone float->u32
  unsigned u = __float_as_uint(f);
  return (u & 0x80000000u) ? ~u : (u | 0x80000000u);
}

// ---------------------------------------------------------------------------
// Separable Gaussian blur, one 16x16 tile per wave, f32 WMMA K-chunks of 4.
// horiz: out = Patch(16 x cols) * G(cols x 16);  vert: out = G'(16 x cols) * Patch
// The banded kernel matrix G[j][n] = g[j-n] is read from a zero-padded LDS tap
// table so every lane issues an unconditional ds_load (no EXEC predication in
// the WMMA loop). H and W are always multiples of 16 (1024 .. 16).
// ---------------------------------------------------------------------------
__global__ void __launch_bounds__(256) blur_wmma(
    const float* __restrict__ src, float* __restrict__ dst,
    int H, int W, float sigma, int k, int horiz) {
  __shared__ float tpad[64];       // zeros, taps at [TPAD, TPAD+k)
  __shared__ float patch[8][656];  // per-wave patch, <= 16*41 floats

  const int tid = threadIdx.x;
  if (tid == 0) {
    for (int i = 0; i < 64; ++i) tpad[i] = 0.0f;
    float sum = 0.0f;
    const float c = (float)(k - 1) * 0.5f;
    const float is2 = 1.0f / (2.0f * sigma * sigma);
    for (int i = 0; i < k; ++i) {
      float d = (float)i - c;
      float g = __expf(-d * d * is2);
      tpad[TPAD + i] = g;
      sum += g;
    }
    float inv = 1.0f / sum;
    for (int i = 0; i < k; ++i) tpad[TPAD + i] *= inv;
  }
  __syncthreads();

  const int lane = tid & 31;
  const int wave = tid >> 5;
  const int tilesX = W >> 4;
  const int tilesY = H >> 4;
  const int t = blockIdx.x * 8 + wave;
  if (t >= tilesX * tilesY) return;  // wave-uniform: EXEC stays all-ones

  const int ty = t / tilesX, tx = t - ty * tilesX;
  const int x0 = tx << 4, y0 = ty << 4;
  const int p = k >> 1;
  const int KC = (15 + k + 3) >> 2;  // number of K=4 chunks
  const int cols = KC << 2;          // padded K extent (<= 40)
  float* pw = patch[wave];

  // --- stage patch into LDS with CDNA5 async loads (per-lane scattered) ---
  const int total = cols * 16;  // elements either orientation
  for (int i = lane; i < total; i += 32) {
    int r, c2, gy, gx;
    if (horiz) {
      r = i / cols; c2 = i - r * cols;
      gy = y0 + r;
      gx = reflecti(x0 - p + c2, W);
    } else {
      r = i >> 4; c2 = i & 15;
      gy = reflecti(y0 - p + r, H);
      gx = x0 + c2;
    }
    const float* ga = src + (size_t)gy * W + gx;
    unsigned la = (unsigned)(size_t)(pw + i);  // LDS byte address (low 32 bits)
    asm volatile("global_load_async_to_lds_b32 %0, %1, off"
                 :: "v"(la), "v"(ga) : "memory");
  }
  asm volatile("s_wait_asynccnt 0" ::: "memory");
  // patch is private to this wave; no cross-wave barrier needed.

  // --- WMMA accumulate: D = A x B + C with K chunks of 4 ---
  v8f acc = {0.f, 0.f, 0.f, 0.f, 0.f, 0.f, 0.f, 0.f};
  const int half = (lane >> 4) << 1;          // 0 for lanes 0-15, 2 for 16-31
  const int mn = lane & 15;
  const float* tb = &tpad[TPAD - mn];         // tb[j] = G[j][mn] (unconditional)
  if (horiz) {
    const float* dr = &pw[mn * cols];         // data row mn
    for (int kc = 0; kc < KC; ++kc) {
      const int j0 = (kc << 2) + half;
      v2f a, b;
      a.x = dr[j0];     a.y = dr[j0 + 1];     // A = data rows
      b.x = tb[j0];     b.y = tb[j0 + 1];     // B = banded kernel matrix
      acc = __builtin_amdgcn_wmma_f32_16x16x4_f32(
          false, a, false, b, (short)0, acc, false, false);
    }
  } else {
    const float* dc = &pw[mn];                // data column mn
    for (int kc = 0; kc < KC; ++kc) {
      const int j0 = (kc << 2) + half;
      v2f a, b;
      a.x = tb[j0];           a.y = tb[j0 + 1];        // A = G'[m][j]=g[j-m]
      b.x = dc[j0 * 16];      b.y = dc[(j0 + 1) * 16]; // B = data rows (K x 16)
      acc = __builtin_amdgcn_wmma_f32_16x16x4_f32(
          false, a, false, b, (short)0, acc, false, false);
    }
  }

  // --- store 16x16 C/D tile (VGPR r: M=r lanes0-15, M=r+8 lanes16-31) ---
  const int mb = (lane < 16) ? 0 : 8;
  float* drow = dst + (size_t)(y0 + mb) * W + (x0 + mn);
  #pragma unroll
  for (int r = 0; r < 8; ++r) drow[(size_t)r * W] = acc[r];
}

// ---------------------------------------------------------------------------
// 2x decimation: next octave base = level-3 of previous octave [::2, ::2]
// ---------------------------------------------------------------------------
__global__ void down_k(const float* __restrict__ src, float* __restrict__ dst,
                       int Hd, int Wd, int Ws) {
  int i = blockIdx.x * blockDim.x + threadIdx.x;
  if (i >= Hd * Wd) return;
  int y = i / Wd, x = i - y * Wd;
  dst[i] = src[(size_t)(2 * y) * Ws + 2 * x];
}

// ---------------------------------------------------------------------------
// 5x5 Hessian-determinant response with edge padding, scaled by sigma^4.
// ---------------------------------------------------------------------------
__global__ void hessian_k(const float* __restrict__ src, float* __restrict__ dst,
                          int H, int W, float sig) {
  int i = blockIdx.x * blockDim.x + threadIdx.x;
  if (i >= H * W) return;
  int y = i / W, x = i - y * W;

  const float GXX[5][5] = {
      {-1, 0, 2, 0, -1}, {-4, 0, 8, 0, -4}, {-6, 0, 12, 0, -6},
      {-4, 0, 8, 0, -4}, {-1, 0, 2, 0, -1}};
  const float GXY[5][5] = {
      {-1, -2, 0, 2, 1}, {-2, -4, 0, 4, 2}, {0, 0, 0, 0, 0},
      {2, 4, 0, -4, -2}, {1, 2, 0, -2, -1}};

  float dxx = 0.f, dxy = 0.f, dyy = 0.f;
  #pragma unroll
  for (int r = 0; r < 5; ++r) {
    int yy = clampi(y + r - 2, 0, H - 1);
    #pragma unroll
    for (int c = 0; c < 5; ++c) {
      int xx = clampi(x + c - 2, 0, W - 1);
      float v = src[(size_t)yy * W + xx];
      dxx += GXX[r][c] * v;
      dyy += GXX[c][r] * v;  // transpose
      dxy += GXY[r][c] * v;
    }
  }
  dxx *= (1.0f / 64.0f);
  dyy *= (1.0f / 64.0f);
  dxy *= (1.0f / 36.0f);
  float s2 = sig * sig;
  dst[i] = (dxx * dyy - dxy * dxy) * s2 * s2;
}

// ---------------------------------------------------------------------------
// 3x3x3 NMS + quadratic refinement. Shared by ranking and re-evaluation.
// ---------------------------------------------------------------------------
__device__ void candidate_eval(const float* __restrict__ resp,
                               int s, int y, int x, int H, int W,
                               float* val, float* cs, float* cx, float* cy) {
  float c[3][3][3];
  #pragma unroll
  for (int ds = 0; ds < 3; ++ds)
    #pragma unroll
    for (int dy = 0; dy < 3; ++dy)
      #pragma unroll
      for (int dx = 0; dx < 3; ++dx) {
        int ss = clampi(s + ds - 1, 0, NLVL - 1);
        int yy = clampi(y + dy - 1, 0, H - 1);
        int xx = clampi(x + dx - 1, 0, W - 1);
        c[ds][dy][dx] = resp[(size_t)ss * PLANE + (size_t)yy * W + xx];
      }
  float ctr = c[1][1][1];

  // strict 26-neighbor max; out-of-volume = -inf (skip)
  float m = NEG_BIG;
  #pragma unroll
  for (int ds = 0; ds < 3; ++ds)
    #pragma unroll
    for (int dy = 0; dy < 3; ++dy)
      #pragma unroll
      for (int dx = 0; dx < 3; ++dx) {
        if (ds == 1 && dy == 1 && dx == 1) continue;
        int ss = s + ds - 1, yy = y + dy - 1, xx = x + dx - 1;
        if (ss >= 0 && ss < NLVL && yy >= 0 && yy < H && xx >= 0 && xx < W)
          m = fmaxf(m, c[ds][dy][dx]);  // in-volume clamped == raw
      }
  bool nms = ctr > m;

  float gx = 0.5f * (c[1][1][2] - c[1][1][0]);
  float gy = 0.5f * (c[1][2][1] - c[1][0][1]);
  float gs = 0.5f * (c[2][1][1] - c[0][1][1]);
  float dxx = c[1][1][2] - 2.f * ctr + c[1][1][0];
  float dyy = c[1][2][1] - 2.f * ctr + c[1][0][1];
  float dss = c[2][1][1] - 2.f * ctr + c[0][1][1];
  float dxy = 0.25f * (c[1][2][2] - c[1][2][0] - c[1][0][2] + c[1][0][0]);
  float dxs = 0.25f * (c[2][1][2] - c[2][1][0] - c[0][1][2] + c[0][1][0]);
  float dys = 0.25f * (c[2][2][1] - c[2][0][1] - c[0][2][1] + c[0][0][1]);

  float a00 = dyy * dss - dys * dys;
  float a01 = dys * dxs - dxy * dss;
  float a02 = dxy * dys - dyy * dxs;
  float a11 = dxx * dss - dxs * dxs;
  float a12 = dxy * dxs - dxx * dys;
  float a22 = dxx * dyy - dxy * dxy;
  float det = dxx * a00 + dxy * a01 + dxs * a02;
  bool solv = fabsf(det) > 1e-7f;
  float inv = solv ? 1.0f / det : 0.0f;
  float ox = -inv * (a00 * gx + a01 * gy + a02 * gs);
  float oy = -inv * (a01 * gx + a11 * gy + a12 * gs);
  float os = -inv * (a02 * gx + a12 * gy + a22 * gs);
  bool big = fmaxf(fmaxf(fabsf(ox), fabsf(oy)), fabsf(os)) > 0.7f;
  bool ok = nms && solv && !big;
  if (!ok) { ox = 0.f; oy = 0.f; os = 0.f; }

  *val = ctr + 0.5f * (gx * ox + gy * oy + gs * os) + (nms ? 10.0f : 0.0f);
  *cs = (float)s + os;
  *cx = (float)x + ox;
  *cy = (float)y + oy;
}

__global__ void zero_buckets_k(unsigned long long* b) {
  int i = blockIdx.x * blockDim.x + threadIdx.x;
  if (i < NFEAT) b[i] = 0ull;
}

// rank all candidates: atomic-max of (key<<32 | idx) into idx%2048 buckets
__global__ void nms_rank_k(const float* __restrict__ resp,
                           unsigned long long* __restrict__ buckets,
                           int H, int W) {
  int N = NLVL * H * W;
  int idx = blockIdx.x * blockDim.x + threadIdx.x;
  if (idx >= N) return;
  int x = idx % W;
  int y = (idx / W) % H;
  int s = idx / (W * H);
  float val, cs, cx, cy;
  candidate_eval(resp, s, y, x, H, W, &val, &cs, &cx, &cy);
  unsigned long long pack =
      ((unsigned long long)fkey(val) << 32) | (unsigned)idx;
  atomicMax(&buckets[idx & (NFEAT - 1)], pack);
}

// re-evaluate the 2048 winners exactly; emit resp*good and laf*px
__global__ void feat_k(const float* __restrict__ resp,
                       const unsigned long long* __restrict__ buckets,
                       float* __restrict__ fR, float* __restrict__ fL,
                       int H, int W, float px) {
  int j = blockIdx.x * blockDim.x + threadIdx.x;
  if (j >= NFEAT) return;
  unsigned long long pack = buckets[j];
  if (pack == 0ull) {  // empty bucket (small octaves) -> never selected
    fR[j] = NEG_BIG;
    #pragma unroll
    for (int q = 0; q < 6; ++q) fL[j * 6 + q] = 0.f;
    return;
  }
  int idx = (int)(pack & 0xffffffffull);
  int x = idx % W;
  int y = (idx / W) % H;
  int s = idx / (W * H);
  float val, cs, cx, cy;
  candidate_eval(resp, s, y, x, H, W, &val, &cs, &cx, &cy);

  float si = fminf(fmaxf(cs, 0.0f), (float)(NLVL - 1));
  int lo = clampi((int)si, 0, NLVL - 2);
  float tt = si - (float)lo;
  float sg = LVL_SIG[lo] + tt * (LVL_SIG[lo + 1] - LVL_SIG[lo]);
  float r = 6.0f * sg;
  bool good = (cx - r >= 0.0f) && (cx + r <= (float)W) &&
              (cy - r >= 0.0f) && (cy + r <= (float)H);
  fR[j] = good ? val : 0.0f;
  fL[j * 6 + 0] = r * px;
  fL[j * 6 + 1] = 0.0f;
  fL[j * 6 + 2] = cx * px;
  fL[j * 6 + 3] = 0.0f;
  fL[j * 6 + 4] = r * px;
  fL[j * 6 + 5] = cy * px;
}

// ---------------------------------------------------------------------------
// Global top-2048: one workgroup, 16K-entry bitonic sort fully in LDS (128KB;
// CDNA5 allows up to 320KB per workgroup). Descending by response.
// ---------------------------------------------------------------------------
__global__ void __launch_bounds__(1024) final_topk(
    const float* __restrict__ fR, const float* __restrict__ fL,
    float* __restrict__ out, int total) {
  extern __shared__ char smem[];
  float* key = (float*)smem;                         // 16384 floats
  int* sidx = (int*)(smem + 16384 * sizeof(float));  // 16384 ints
  const int SORTN = 16384;
  const int tid = threadIdx.x;

  for (int i = tid; i < SORTN; i += 1024) {
    key[i] = (i < total) ? fR[i] : NEG_BIG;
    sidx[i] = i;
  }
  __syncthreads();

  for (unsigned k2 = 2; k2 <= (unsigned)SORTN; k2 <<= 1) {
    for (unsigned j = k2 >> 1; j > 0; j >>= 1) {
      for (unsigned i = tid; i < (unsigned)SORTN; i += 1024) {
        unsigned ixj = i ^ j;
        if (ixj > i) {
          bool desc = ((i & k2) == 0);
          float a = key[i], b = key[ixj];
          bool sw = desc ? (a < b) : (a > b);
          if (sw) {
            key[i] = b; key[ixj] = a;
            int ti = sidx[i]; sidx[i] = sidx[ixj]; sidx[ixj] = ti;
          }
        }
      }
      __syncthreads();
    }
  }

  // out = lafs (2048*2*3) then resps (2048)
  for (int j = tid; j < NFEAT; j += 1024) {
    int s = sidx[j];
    #pragma unroll
    for (int q = 0; q < 6; ++q) out[j * 6 + q] = fL[s * 6 + q];
    out[NFEAT * 6 + j] = key[j];
  }
}

// ---------------------------------------------------------------------------
// Host orchestration (graph-capture safe: launches only).
// Workspace layout (needs ~53MB):
//   ss[6 planes] | tmp[1 plane] | resp[6 planes] | buckets[2048 u64] |
//   featR[7*2048 f32] | featL[7*2048*6 f32]
// ---------------------------------------------------------------------------
extern "C" void kernel_launch(void* const* d_in, const int* in_sizes, int n_in,
                              void* d_out, int out_size, void* d_ws,
                              size_t ws_size, hipStream_t stream) {
  (void)in_sizes; (void)n_in; (void)out_size; (void)ws_size;
  const float* img = (const float*)d_in[0];
  float* ss = (float*)d_ws;
  float* tmp = ss + (size_t)NLVL * PLANE;
  float* respv = tmp + (size_t)PLANE;
  unsigned long long* buckets =
      (unsigned long long*)(respv + (size_t)NLVL * PLANE);
  float* featR = (float*)(buckets + NFEAT);
  float* featL = featR + (size_t)NOCT * NFEAT;

  // compile-time-known sigmas / kernel sizes
  const double STEP = 1.2599210498948732;           // 2^(1/3)
  const double bf = sqrt(STEP * STEP - 1.0);        // 0.766421...
  double lvl[NLVL];
  for (int i = 0; i < NLVL; ++i) lvl[i] = 1.6 * pow(STEP, (double)i);
  double bsig[NLVL];
  bsig[0] = sqrt(1.6 * 1.6 - 0.25);                 // initial blur
  for (int i = 1; i < NLVL; ++i) bsig[i] = lvl[i - 1] * bf;
  int ks[NLVL];
  for (int i = 0; i < NLVL; ++i) {
    int k = (int)(8.0 * bsig[i] + 1.0);
    ks[i] = (k & 1) ? k : k + 1;                    // 13,11,13,17,21,25
  }

  int H = 1024, W = 1024;
  float px = 1.0f;
  for (int o = 0; o < NOCT; ++o) {
    int tiles = (H >> 4) * (W >> 4);
    int blurBlks = (tiles + 7) / 8;

    if (o == 0) {
      blur_wmma<<<blurBlks, 256, 0, stream>>>(img, tmp, H, W,
                                              (float)bsig[0], ks[0], 1);
      blur_wmma<<<blurBlks, 256, 0, stream>>>(tmp, ss, H, W,
                                              (float)bsig[0], ks[0], 0);
    } else {
      down_k<<<(H * W + 255) / 256, 256, 0, stream>>>(
          ss + (size_t)3 * PLANE, ss, H, W, 2 * W);
    }
    for (int i = 1; i < NLVL; ++i) {
      blur_wmma<<<blurBlks, 256, 0, stream>>>(
          ss + (size_t)(i - 1) * PLANE, tmp, H, W, (float)bsig[i], ks[i], 1);
      blur_wmma<<<blurBlks, 256, 0, stream>>>(
          tmp, ss + (size_t)i * PLANE, H, W, (float)bsig[i], ks[i], 0);
    }
    for (int i = 0; i < NLVL; ++i)
      hessian_k<<<(H * W + 255) / 256, 256, 0, stream>>>(
          ss + (size_t)i * PLANE, respv + (size_t)i * PLANE, H, W,
          (float)lvl[i]);

    zero_buckets_k<<<NFEAT / 256, 256, 0, stream>>>(buckets);
    int N = NLVL * H * W;
    nms_rank_k<<<(N + 255) / 256, 256, 0, stream>>>(respv, buckets, H, W);
    feat_k<<<NFEAT / 256, 256, 0, stream>>>(
        respv, buckets, featR + (size_t)o * NFEAT,
        featL + (size_t)o * NFEAT * 6, H, W, px);

    px *= 2.0f;
    H >>= 1;
    W >>= 1;
  }

  final_topk<<<1, 1024, 16384 * 8, stream>>>(featR, featL, (float*)d_out,
                                             NOCT * NFEAT);
}